// Linear_46497315947082
// MI455X (gfx1250) — compile-verified
//
#include <hip/hip_runtime.h>

typedef __attribute__((ext_vector_type(2))) float v2f;
typedef __attribute__((ext_vector_type(8))) float v8f;

#define ROWLEN 1184            // floats per row of x / out
#define TPW 5                  // batch tiles (16 rows each) per wave
#define GROUPS 1250            // 6250 batch tiles / TPW (exact)
#define NSTRIP 37              // sum over segs of d * (mv/32)  (v-tile PAIRS)
#define TOTAL_WAVES (NSTRIP * GROUPS)
#define WAVES_PER_BLOCK 8
#define BLOCK_THREADS (WAVES_PER_BLOCK * 32)

// One strip = (segment, ir-component, pair of 16-wide v-tiles).
// A fragments are shared by both v-tiles -> 2 WMMAs per A load pair.
template<int MU, int D, int XO, int WO>
__device__ __forceinline__ void strip_body(const float* __restrict__ x,
                                           const float* __restrict__ w,
                                           float* __restrict__ out,
                                           float norm, int comp, int vp,
                                           int brow0, int m, int half, int kh)
{
    constexpr int MV = MU;

    const float* arow[TPW];
#pragma unroll
    for (int t = 0; t < TPW; ++t)
        arow[t] = x + (size_t)(brow0 + t * 16 + m) * ROWLEN + XO + comp;

    // B columns: v-tile pair base = (2*vp)*16; lane supplies col n = lane&15.
    const float* wcol = w + WO + vp * 32 + m;

    v8f c0[TPW] = {};
    v8f c1[TPW] = {};

#pragma unroll 2
    for (int k0 = 0; k0 < MU; k0 += 4) {
        v2f bf0, bf1;
        bf0.x = wcol[(size_t)(k0 + kh)     * MV]      * norm;
        bf0.y = wcol[(size_t)(k0 + kh + 1) * MV]      * norm;
        bf1.x = wcol[(size_t)(k0 + kh)     * MV + 16] * norm;
        bf1.y = wcol[(size_t)(k0 + kh + 1) * MV + 16] * norm;
#pragma unroll
        for (int t = 0; t < TPW; ++t) {
            v2f a;
            if constexpr (D == 1) {
                // contiguous K elements -> single global_load_b64
                a = *(const v2f*)(arow[t] + (k0 + kh));
            } else {
                a.x = arow[t][(size_t)(k0 + kh)     * D];
                a.y = arow[t][(size_t)(k0 + kh + 1) * D];
            }
            c0[t] = __builtin_amdgcn_wmma_f32_16x16x4_f32(
                        false, a, false, bf0, (short)0, c0[t], false, false);
            c1[t] = __builtin_amdgcn_wmma_f32_16x16x4_f32(
                        false, a, false, bf1, (short)0, c1[t], false, false);
        }
    }

    // C/D layout: VGPR j -> row (half*8 + j), col n = lane&15.
    const int col0 = XO + ((vp * 2 + 0) * 16 + m) * D + comp;
    const int col1 = XO + ((vp * 2 + 1) * 16 + m) * D + comp;
#pragma unroll
    for (int t = 0; t < TPW; ++t) {
        float* orow = out + (size_t)(brow0 + t * 16 + half * 8) * ROWLEN;
#pragma unroll
        for (int j = 0; j < 8; ++j) {
            orow[(size_t)j * ROWLEN + col0] = c0[t][j];
            orow[(size_t)j * ROWLEN + col1] = c1[t][j];
        }
    }
}

__global__ __launch_bounds__(BLOCK_THREADS)
void seglinear_wmma_f32(const float* __restrict__ x,
                        const float* __restrict__ w,
                        float* __restrict__ out)
{
    const int lane        = threadIdx.x & 31;
    const int waveInBlock = threadIdx.x >> 5;
    const int waveId      = blockIdx.x * WAVES_PER_BLOCK + waveInBlock;
    if (waveId >= TOTAL_WAVES) return;        // wave-uniform guard

    // strip-major within a batch group: waves reading the same x rows are
    // temporally adjacent -> x re-reads across v-tiles hit L2, not HBM.
    const int strip = waveId % NSTRIP;
    const int group = waveId / NSTRIP;

    // decode strip -> (segment, ir-component, v-tile pair); wave-uniform
    int seg, comp, vp;
    {
        int s = strip;
        if (s < 8)       { seg = 0; comp = 0;      vp = s;     }
        else if (s < 20) { s -= 8;  seg = 1; comp = s >> 2; vp = s & 3; }
        else if (s < 30) { s -= 20; seg = 2; comp = s >> 1; vp = s & 1; }
        else             { s -= 30; seg = 3; comp = s;      vp = 0;     }
    }

    // 32-bit WMMA fragment lane roles (ISA 7.12.2):
    //   A 16x4 : lane L holds row M = L&15;  V0 -> K = 2*(L>>4), V1 -> K+1
    //   B 4x16 : lane L holds col N = L&15;  same K split as A
    //   C/D    : VGPR j, lanes 0-15 -> (M=j, N=lane); lanes 16-31 -> (M=j+8)
    const int m    = lane & 15;
    const int half = lane >> 4;
    const int kh   = half * 2;

    const int brow0 = group * (TPW * 16);

    switch (seg) {
        case 0:
            strip_body<256, 1, 0, 0>(x, w, out, 0.0625f,
                                     comp, vp, brow0, m, half, kh);
            break;
        case 1:
            strip_body<128, 3, 256, 65536>(x, w, out, 0.08838834764831845f,
                                           comp, vp, brow0, m, half, kh);
            break;
        case 2:
            strip_body<64, 5, 640, 81920>(x, w, out, 0.125f,
                                          comp, vp, brow0, m, half, kh);
            break;
        default:
            strip_body<32, 7, 960, 86016>(x, w, out, 0.17677669529663687f,
                                          comp, vp, brow0, m, half, kh);
            break;
    }
}

extern "C" void kernel_launch(void* const* d_in, const int* in_sizes, int n_in,
                              void* d_out, int out_size, void* d_ws, size_t ws_size,
                              hipStream_t stream)
{
    const float* x = (const float*)d_in[0];   // [100000, 1184] fp32
    const float* w = (const float*)d_in[1];   // [1, 87040]    fp32
    float* out     = (float*)d_out;           // [100000, 1184] fp32

    const int nblocks = (TOTAL_WAVES + WAVES_PER_BLOCK - 1) / WAVES_PER_BLOCK;
    seglinear_wmma_f32<<<nblocks, BLOCK_THREADS, 0, stream>>>(x, w, out);
}